// FeatureViewEncoder_26139170963542
// MI455X (gfx1250) — compile-verified
//
#include <hip/hip_runtime.h>

typedef __attribute__((ext_vector_type(2))) float v2f;
typedef __attribute__((ext_vector_type(8))) float v8f;

#define B_   128
#define V_   200
#define F_   64
#define H_   128
#define L_   198            // V - K + 1
#define LT_  13             // ceil(L/16)
#define HT_  8              // H/16
#define NEGINF_  (-1.0e9f)
#define MASKPEN_ (-2.0e9f)  // pushed below NEGINF so masked entries never win

__global__ __launch_bounds__(256)
void fve_wmma_kernel(const float* __restrict__ x,      // [B,V,F]
                     const int*   __restrict__ vmask,  // [B,1,L]
                     const float* __restrict__ cw,     // [F,H,3]
                     const float* __restrict__ cb,     // [F,H]
                     float*       __restrict__ out)    // [B,F,H]
{
    const int tid  = threadIdx.x;
    const int lane = tid & 31;
    const int half = lane >> 4;     // 0: lanes 0-15, 1: lanes 16-31
    const int ln   = lane & 15;
    const int wid  = (int)((blockIdx.x * blockDim.x + tid) >> 5); // 0..8191
    const int b = wid >> 6;         // / F_
    const int f = wid & 63;         // % F_

    const float* xb = x + b * (V_ * F_) + f;   // &x[b, 0, f], row stride F_
    const int*   mb = vmask + b * L_;          // &vmask[b, 0]

    // ---- preload B-matrix (weights) for all 8 h-tiles, K padded 3->4 ----
    // f32 4x16 B layout (mirror of documented 16x4 A layout):
    //   VGPR0: lanes 0-15 -> K=0,N=ln ; lanes 16-31 -> K=2,N=ln
    //   VGPR1: lanes 0-15 -> K=1,N=ln ; lanes 16-31 -> K=3 (zero pad)
    float wb0[HT_], wb1[HT_];
#pragma unroll
    for (int ht = 0; ht < HT_; ++ht) {
        const float* wp = cw + (f * H_ + ht * 16 + ln) * 3;
        wb0[ht] = half ? wp[2] : wp[0];
        wb1[ht] = half ? 0.0f  : wp[1];
    }

    float acc[HT_];
#pragma unroll
    for (int ht = 0; ht < HT_; ++ht) acc[ht] = -3.0e9f;

#pragma unroll 1
    for (int lt = 0; lt < LT_; ++lt) {
        const int base = lt * 16;

        // ---- A-matrix: A[m][k] = x[b, base+m+k, f]  (clamped; pad rows poisoned via C)
        int r0 = base + ln + (half ? 2 : 0);   // taps k=0 (lo half) / k=2 (hi half)
        int r1 = base + ln + 1;                // tap  k=1 (lo half only)
        r0 = (r0 > V_ - 1) ? (V_ - 1) : r0;    // branch-free clamp (v_min)
        r1 = (r1 > V_ - 1) ? (V_ - 1) : r1;
        v2f A;
        A.x = xb[r0 * F_];
        A.y = half ? 0.0f : xb[r1 * F_];

        // ---- mask word for this l-tile -> C-matrix addend (shared by all h-tiles)
        // branch-free: clamped address, then zero out-of-range lanes with a select
        const int mi  = base + ln;
        const int mic = (mi > L_ - 1) ? (L_ - 1) : mi;
        int mv = mb[mic];
        mv = (mi < L_) ? mv : 0;
        const unsigned int bm = (unsigned int)__ballot(mv != 0); // bit i = mask[base+(i&15)]
        const unsigned int sh = bm >> (half * 8);                // this half's 8 rows
        v8f C;
#pragma unroll
        for (int r = 0; r < 8; ++r)
            C[r] = ((sh >> r) & 1u) ? 0.0f : MASKPEN_;

        // ---- 8 WMMAs: D = A(16x4) x B(4x16) + C, then max-reduce over M (=l)
        // sequential chain -> compiler fuses into v_max3_num_f32
#pragma unroll
        for (int ht = 0; ht < HT_; ++ht) {
            v2f Bm; Bm.x = wb0[ht]; Bm.y = wb1[ht];
            v8f D = __builtin_amdgcn_wmma_f32_16x16x4_f32(
                false, A, false, Bm, (short)0, C, false, false);
            float m = D[0];
#pragma unroll
            for (int r = 1; r < 8; ++r) m = fmaxf(m, D[r]);
            acc[ht] = fmaxf(acc[ht], m);
        }
    }

    // ---- combine lane-halves (M 0-7 vs 8-15), add bias, clamp, store ----
#pragma unroll
    for (int ht = 0; ht < HT_; ++ht) {
        const float other = __shfl_xor(acc[ht], 16, 32);
        const float tot   = fmaxf(acc[ht], other);
        const float bias  = cb[f * H_ + ht * 16 + ln];
        const float res   = fmaxf(tot + bias, NEGINF_);
        if (half == 0)
            out[(b * F_ + f) * H_ + ht * 16 + ln] = res;
    }
}

extern "C" void kernel_launch(void* const* d_in, const int* in_sizes, int n_in,
                              void* d_out, int out_size, void* d_ws, size_t ws_size,
                              hipStream_t stream) {
    const float* x  = (const float*)d_in[0];   // input_visit [128,200,64] f32
    const int*   vm = (const int*)  d_in[1];   // visit_mask  [128,1,198] i32
    const float* cw = (const float*)d_in[2];   // conv_w      [64,128,3]  f32
    const float* cb = (const float*)d_in[3];   // conv_b      [64,128]    f32
    float* out = (float*)d_out;                // [128,64,128] f32

    // 8192 waves (one per (b,f)) = 1024 blocks x 8 waves
    fve_wmma_kernel<<<dim3(1024), dim3(256), 0, stream>>>(x, vm, cw, cb, out);
}